// WaveletDWTLayer_3152505995603
// MI455X (gfx1250) — compile-verified
//
#include <hip/hip_runtime.h>

typedef float v4f __attribute__((ext_vector_type(4)));

#define Bb      2048
#define Nn      16384
#define Ff      8
#define Ll      8195                    // (N + F - 1) / 2
#define TL      1024                    // outputs per block tile
#define TILE_X  (2 * TL + 8)            // 2056 ext elements staged in LDS
#define NTILES  ((Ll + TL - 1) / TL)    // 9
#define THREADS 256

// ---------------------------------------------------------------------------
// DWT kernel: one block = one (batch row, output tile).
// Stages the symmetric-extended window into LDS with CDNA5 async loads
// (tracked by ASYNCcnt), then computes cA/cD with register-held taps.
// ---------------------------------------------------------------------------
__global__ __launch_bounds__(THREADS) void dwt_kernel(
    const float* __restrict__ x1, const float* __restrict__ w,
    float* __restrict__ outA, float* __restrict__ outD)
{
    __shared__ float tile[TILE_X];

    const int t  = threadIdx.x;
    const int b  = blockIdx.x / NTILES;
    const int tb = blockIdx.x % NTILES;
    const int l0 = tb * TL;
    const float* src = x1 + (long long)b * Nn;

    // --- filter taps from weights (uniform scalar loads, hits K$) ---------
    float wr[Ff];
#pragma unroll
    for (int k = 0; k < Ff; ++k) wr[k] = w[k];
    float s = 0.f;
#pragma unroll
    for (int k = 0; k < Ff; ++k) s += wr[k];
    const float scale = 1.4142135623730951f / s;   // sqrt(2)/sum(w)
    float rl[Ff], rh[Ff];
#pragma unroll
    for (int k = 0; k < Ff; ++k) rl[k] = wr[k] * scale;        // rec_lo
#pragma unroll
    for (int k = 0; k < Ff; ++k) rh[k] = (k & 1) ? -rl[7 - k] : rl[7 - k]; // rec_hi

    // --- async stage of ext[j0 .. j0+TILE_X) into LDS ---------------------
    const int j0 = 2 * l0;
    const unsigned lds_base =
        (unsigned)(unsigned long long)(__attribute__((address_space(3))) void*)tile;

#pragma unroll
    for (int it = 0; it < (TILE_X + THREADS - 1) / THREADS; ++it) {
        int i = t + it * THREADS;
        if (i < TILE_X) {
            int j = j0 + i;
            j = (j > Nn + 12) ? (Nn + 12) : j;               // clamp tail tile
            int gx = (j < 6) ? (5 - j)
                             : ((j <= Nn + 5) ? (j - 6) : (2 * Nn + 5 - j));
            unsigned lds_addr = lds_base + (unsigned)i * 4u;
            unsigned long long gaddr =
                (unsigned long long)(const void*)(src + gx);
            asm volatile("global_load_async_to_lds_b32 %0, %1, off"
                         :: "v"(lds_addr), "v"(gaddr) : "memory");
        }
    }
#if defined(__has_builtin) && __has_builtin(__builtin_amdgcn_s_wait_asynccnt)
    __builtin_amdgcn_s_wait_asynccnt(0);
#else
    asm volatile("s_wait_asynccnt 0" ::: "memory");
#endif
    __syncthreads();

    // --- compute: 4 output pairs per thread; wave32 stride-2 LDS reads
    //     touch 32 distinct banks (of 64) -> conflict free ------------------
#pragma unroll
    for (int sidx = 0; sidx < TL / THREADS; ++sidx) {
        int lt = t + sidx * THREADS;
        int l  = l0 + lt;
        if (l < Ll) {
            const float* p = &tile[2 * lt];
            float a = 0.f, d = 0.f;
#pragma unroll
            for (int k = 0; k < Ff; ++k) {
                float v = p[k];
                a = fmaf(rl[k], v, a);
                d = fmaf(rh[k], v, d);
            }
            long long o = (long long)b * Ll + l;
            __builtin_nontemporal_store(a, outA + o);
            __builtin_nontemporal_store(d, outD + o);
        }
    }
}

// ---------------------------------------------------------------------------
// x3 pass-through: pure streaming copy, b128 + non-temporal on both sides so
// it doesn't evict x1/cA/cD lines from L2.
// ---------------------------------------------------------------------------
__global__ __launch_bounds__(256) void copy_kernel(
    const v4f* __restrict__ src, v4f* __restrict__ dst, long long n4)
{
    long long i      = (long long)blockIdx.x * blockDim.x + threadIdx.x;
    long long stride = (long long)gridDim.x * blockDim.x;
    for (; i < n4; i += stride) {
        v4f v = __builtin_nontemporal_load(src + i);
        __builtin_nontemporal_store(v, dst + i);
    }
}

extern "C" void kernel_launch(void* const* d_in, const int* in_sizes, int n_in,
                              void* d_out, int out_size, void* d_ws, size_t ws_size,
                              hipStream_t stream) {
    const float* x1  = (const float*)d_in[0];
    // d_in[1] (x2) is unused by the reference.
    const float* x3  = (const float*)d_in[2];
    const float* wts = (const float*)d_in[3];

    float* out   = (float*)d_out;
    float* outA  = out;                              // cA : B*L
    float* outD  = out + (long long)Bb * Ll;         // cD : B*L
    float* outX3 = out + 2LL * Bb * Ll;              // x3 : B*N (16B aligned)

    dwt_kernel<<<dim3(Bb * NTILES), dim3(THREADS), 0, stream>>>(x1, wts, outA, outD);

    long long n4 = (long long)Bb * Nn / 4;           // 8,388,608 float4s
    copy_kernel<<<dim3(8192), dim3(256), 0, stream>>>(
        (const v4f*)x3, (v4f*)outX3, n4);
}